// Decoder_28991029248558
// MI455X (gfx1250) — compile-verified
//
#include <hip/hip_runtime.h>
#include <math.h>

// Problem dims (fixed by reference)
#define B_   64
#define TD_  256
#define TE_  256
#define H_   512
#define D_   256
#define ZN_  2048   // 4*H

typedef __attribute__((ext_vector_type(2))) float v2f;
typedef __attribute__((ext_vector_type(8))) float v8f;

// D = A(16x4,f32) * B(4x16,f32) + C(16x16,f32), wave32.
// Layout (ISA 7.12.2): lane = l, ml = l&15, kh = l>>4.
//   A: a.x = A[ml][k0+2*kh], a.y = A[ml][k0+2*kh+1]
//   B: b.x = B[k0+2*kh][n0+ml], b.y = B[k0+2*kh+1][n0+ml]
//   C/D: c[r] = C[m0 + r + 8*kh][n0 + ml]
__device__ __forceinline__ v8f wmma_f32(v2f a, v2f b, v8f c) {
  return __builtin_amdgcn_wmma_f32_16x16x4_f32(
      /*neg_a=*/false, a, /*neg_b=*/false, b,
      /*c_mod=*/(short)0, c, /*reuse_a=*/false, /*reuse_b=*/false);
}

// ---------------------------------------------------------------------------
// Kernel 1: XW[m][n] = E[tok[m]][:] @ W[:, n] + bias[n], m = b*TD + t.
// M=16384, K=256, N=2048. One wave per 16x16 tile, 8 waves per block.
// ---------------------------------------------------------------------------
__global__ __launch_bounds__(256)
void xw_gemm_kernel(const int* __restrict__ tok, const float* __restrict__ E,
                    const float* __restrict__ W, const float* __restrict__ bias,
                    float* __restrict__ XW) {
  const int lane = threadIdx.x & 31;
  const int wave = threadIdx.x >> 5;
  const int tile = blockIdx.x * 8 + wave;       // 131072 tiles total
  const int tile_n = tile & 127;                // N/16 = 128
  const int tile_m = tile >> 7;                 // M/16 = 1024
  const int ml = lane & 15;
  const int kh = lane >> 4;
  const int m0 = tile_m * 16;
  const int n0 = tile_n * 16;

  const int t = tok[m0 + ml];                   // per-lane row gather
  const float* arow = E + (long)t * D_;
  const float* bcol = W + n0 + ml;

  v8f acc = {};
#pragma unroll 8
  for (int k0 = 0; k0 < D_; k0 += 4) {
    v2f a, b;
    a.x = arow[k0 + 2 * kh];
    a.y = arow[k0 + 2 * kh + 1];
    b.x = bcol[(long)(k0 + 2 * kh) * ZN_];
    b.y = bcol[(long)(k0 + 2 * kh + 1) * ZN_];
    acc = wmma_f32(a, b, acc);
  }
  const float bv = bias[n0 + ml];
  float* outp = XW + (long)m0 * ZN_ + n0 + ml;
#pragma unroll
  for (int r = 0; r < 8; ++r)
    outp[(long)(r + 8 * kh) * ZN_] = acc[r] + bv;
}

// ---------------------------------------------------------------------------
// Kernel 2: one LSTM timestep. grid = H/16 = 32 blocks, 1024 threads (32 waves).
// Block owns hidden cols [j0, j0+16) across all 4 gates and all 64 batch rows.
// wave = (k_half, gate, m_tile): K=512 split across 2 waves -> each wave runs
// only a 64-deep serial WMMA chain; partials summed from two LDS planes.
// h double-buffered (h_in -> h_out), c updated in place (per-column ownership).
// ---------------------------------------------------------------------------
__global__ __launch_bounds__(1024)
void lstm_step_kernel(const float* __restrict__ XW, const float* __restrict__ U,
                      const int* __restrict__ tok,
                      const float* __restrict__ h_in, float* __restrict__ h_out,
                      float* __restrict__ cbuf, float* __restrict__ dec_out,
                      const int t) {
  __shared__ float zs[2][64][65];               // [k_half][batch row][gate*16+col]

  const int lane = threadIdx.x & 31;
  const int wave = threadIdx.x >> 5;            // 0..31
  const int ks   = wave >> 4;                   // K half: 0 or 1
  const int g    = (wave >> 2) & 3;             // gate: i,f,g,o
  const int mt   = wave & 3;                    // batch 16-row tile
  const int ml   = lane & 15;
  const int kh   = lane >> 4;
  const int m0   = mt * 16;
  const int j0   = blockIdx.x * 16;             // hidden column group
  const int zc0  = g * H_ + j0;                 // column base in 2048-wide z

  // Overlap the post-GEMM XW read with the GEMM: prefetch this block's slice
  // (lowers to global_prefetch_b8 on gfx1250).
  {
    const int prow = threadIdx.x >> 4;          // 0..63 batch row
    const int pcol = threadIdx.x & 15;
    const float* p = XW + ((long)prow * TD_ + t) * ZN_ + j0 + pcol;
    __builtin_prefetch(p,          0, 1);
    __builtin_prefetch(p + H_,     0, 1);
    __builtin_prefetch(p + 2 * H_, 0, 1);
    __builtin_prefetch(p + 3 * H_, 0, 1);
  }

  const float* arow = h_in + (long)(m0 + ml) * H_;
  const float* bcol = U + zc0 + ml;
  const int kbeg = ks * (H_ / 2);

  v8f acc = {};
#pragma unroll 4
  for (int kk = 0; kk < H_ / 2; kk += 4) {
    const int k0 = kbeg + kk;
    v2f a, b;
    a.x = arow[k0 + 2 * kh];
    a.y = arow[k0 + 2 * kh + 1];
    b.x = bcol[(long)(k0 + 2 * kh) * ZN_];
    b.y = bcol[(long)(k0 + 2 * kh + 1) * ZN_];
    acc = wmma_f32(a, b, acc);
  }
#pragma unroll
  for (int r = 0; r < 8; ++r)
    zs[ks][m0 + r + 8 * kh][g * 16 + ml] = acc[r];
  __syncthreads();

  // 64 rows x 16 cols = 1024 elements -> exactly one per thread.
  {
    const int row = threadIdx.x >> 4;
    const int col = threadIdx.x & 15;
    const int jh  = j0 + col;
    const long xb = ((long)row * TD_ + t) * ZN_ + jh;
    const float iz = zs[0][row][col]      + zs[1][row][col]      + XW[xb];
    const float fz = zs[0][row][16 + col] + zs[1][row][16 + col] + XW[xb + H_];
    const float gz = zs[0][row][32 + col] + zs[1][row][32 + col] + XW[xb + 2 * H_];
    const float oz = zs[0][row][48 + col] + zs[1][row][48 + col] + XW[xb + 3 * H_];
    const float c_old = cbuf[row * H_ + jh];
    const float ig = 1.f / (1.f + __expf(-iz));
    const float fg = 1.f / (1.f + __expf(-fz));
    const float og = 1.f / (1.f + __expf(-oz));
    const float gg = tanhf(gz);
    const float c_new = fg * c_old + ig * gg;
    const float h_new = og * tanhf(c_new);
    const bool  msk   = tok[row * TD_ + t] != 0;
    const float h2 = msk ? h_new : h_in[row * H_ + jh];
    const float c2 = msk ? c_new : c_old;
    cbuf[row * H_ + jh]  = c2;
    h_out[row * H_ + jh] = h2;
    dec_out[((long)row * TD_ + t) * H_ + jh] = h2;
  }
}

// ---------------------------------------------------------------------------
// Kernel 3: fused attention per (batch, 16-query tile). 256 threads (8 waves).
// S = Q @ Enc^T (16x256, K=512) -> LDS; softmax; ctx = P @ Enc (16x512, K=256).
// Masked rows of ctx written as zero. Scores never touch HBM.
// ---------------------------------------------------------------------------
__global__ __launch_bounds__(256)
void attn_kernel(const float* __restrict__ dec_out, const float* __restrict__ enc,
                 const int* __restrict__ tok, float* __restrict__ ctx) {
  __shared__ float S[16 * 260];                 // padded stride: 260 % 64 == 4
  __shared__ float red[16][17];
  __shared__ float rowstat[16];

  const int b  = blockIdx.x >> 4;
  const int q0 = (blockIdx.x & 15) * 16;
  const int lane = threadIdx.x & 31;
  const int wave = threadIdx.x >> 5;            // 0..7
  const int ml = lane & 15;
  const int kh = lane >> 4;

  const float* Q  = dec_out + ((long)b * TD_ + q0) * H_;
  const float* Eb = enc + (long)b * TE_ * H_;

  // Stage 1: scores. Each wave computes two 16x16 TE tiles.
  for (int i = 0; i < 2; ++i) {
    const int te0 = (wave * 2 + i) * 16;
    const float* arow = Q + (long)ml * H_;
    const float* brow = Eb + (long)(te0 + ml) * H_;   // B[k][n] = Enc[n][k]
    v8f acc = {};
#pragma unroll 4
    for (int k0 = 0; k0 < H_; k0 += 4) {
      v2f a, bf;
      a.x  = arow[k0 + 2 * kh];  a.y  = arow[k0 + 2 * kh + 1];
      bf.x = brow[k0 + 2 * kh];  bf.y = brow[k0 + 2 * kh + 1];
      acc = wmma_f32(a, bf, acc);
    }
#pragma unroll
    for (int r = 0; r < 8; ++r)
      S[(r + 8 * kh) * 260 + te0 + ml] = acc[r];
  }
  __syncthreads();

  // Softmax over 256 cols per row; 16 threads per row, 16 elements each.
  const int r  = threadIdx.x >> 4;
  const int cg = threadIdx.x & 15;
  float lm = -3.4e38f;
  for (int i = 0; i < 16; ++i) lm = fmaxf(lm, S[r * 260 + cg + 16 * i]);
  red[r][cg] = lm;
  __syncthreads();
  if (cg == 0) {
    float m = red[r][0];
    for (int i = 1; i < 16; ++i) m = fmaxf(m, red[r][i]);
    rowstat[r] = m;
  }
  __syncthreads();
  const float rmax = rowstat[r];
  float ls = 0.f;
  for (int i = 0; i < 16; ++i) {
    const float e = __expf(S[r * 260 + cg + 16 * i] - rmax);
    S[r * 260 + cg + 16 * i] = e;
    ls += e;
  }
  __syncthreads();
  red[r][cg] = ls;
  __syncthreads();
  if (cg == 0) {
    float s = 0.f;
    for (int i = 0; i < 16; ++i) s += red[r][i];
    rowstat[r] = 1.f / s;
  }
  __syncthreads();
  const float rinv = rowstat[r];
  for (int i = 0; i < 16; ++i) S[r * 260 + cg + 16 * i] *= rinv;
  __syncthreads();

  // Stage 2: ctx = P @ Enc. Each wave computes four 16x16 N tiles.
  for (int i = 0; i < 4; ++i) {
    const int n0 = (wave * 4 + i) * 16;
    v8f acc = {};
#pragma unroll 4
    for (int k0 = 0; k0 < TE_; k0 += 4) {
      v2f a, bf;
      a.x  = S[ml * 260 + k0 + 2 * kh];
      a.y  = S[ml * 260 + k0 + 2 * kh + 1];
      bf.x = Eb[(long)(k0 + 2 * kh) * H_ + n0 + ml];
      bf.y = Eb[(long)(k0 + 2 * kh + 1) * H_ + n0 + ml];
      acc = wmma_f32(a, bf, acc);
    }
#pragma unroll
    for (int rr = 0; rr < 8; ++rr) {
      const int q = q0 + rr + 8 * kh;
      const bool msk = tok[b * TD_ + q] != 0;
      ctx[((long)b * TD_ + q) * H_ + n0 + ml] = msk ? acc[rr] : 0.f;
    }
  }
}

// ---------------------------------------------------------------------------
extern "C" void kernel_launch(void* const* d_in, const int* in_sizes, int n_in,
                              void* d_out, int out_size, void* d_ws, size_t ws_size,
                              hipStream_t stream) {
  (void)in_sizes; (void)n_in; (void)out_size; (void)ws_size;
  const float* enc  = (const float*)d_in[0];   // (B, TE, H)
  const int*   tok  = (const int*)d_in[1];     // (B, TD)
  const float* h0   = (const float*)d_in[2];   // (B, H)
  const float* c0   = (const float*)d_in[3];   // (B, H)
  const float* E    = (const float*)d_in[4];   // (V, D)
  const float* W    = (const float*)d_in[5];   // (D, 4H)
  const float* U    = (const float*)d_in[6];   // (H, 4H)
  const float* bias = (const float*)d_in[7];   // (4H,)
  float* out = (float*)d_out;                  // ctx | hT | cT

  float* XW    = (float*)d_ws;                          // 16384 x 2048
  float* dec   = XW + (size_t)B_ * TD_ * ZN_;           // 16384 x 512
  float* hbuf0 = dec + (size_t)B_ * TD_ * H_;
  float* hbuf1 = hbuf0 + (size_t)B_ * H_;
  float* cbuf  = hbuf1 + (size_t)B_ * H_;

  hipMemcpyAsync(hbuf0, h0, (size_t)B_ * H_ * sizeof(float),
                 hipMemcpyDeviceToDevice, stream);
  hipMemcpyAsync(cbuf, c0, (size_t)B_ * H_ * sizeof(float),
                 hipMemcpyDeviceToDevice, stream);

  // Parallel pre-GEMM: XW = gather(E, tok) @ W + b
  xw_gemm_kernel<<<16384, 256, 0, stream>>>(tok, E, W, bias, XW);

  // Sequential recurrence: one small WMMA kernel per timestep (graph-replayed)
  for (int t = 0; t < TD_; ++t) {
    const float* hin = (t & 1) ? hbuf1 : hbuf0;
    float*      hout = (t & 1) ? hbuf0 : hbuf1;
    lstm_step_kernel<<<H_ / 16, 1024, 0, stream>>>(XW, U, tok, hin, hout, cbuf,
                                                   dec, t);
  }
  // TD_ is even -> final h state lands in hbuf0

  // Fused attention straight into d_out
  attn_kernel<<<B_ * 16, 256, 0, stream>>>(dec, enc, tok, out);

  // hT, cT appended after ctx
  hipMemcpyAsync(out + (size_t)B_ * TD_ * H_, hbuf0,
                 (size_t)B_ * H_ * sizeof(float), hipMemcpyDeviceToDevice, stream);
  hipMemcpyAsync(out + (size_t)B_ * TD_ * H_ + (size_t)B_ * H_, cbuf,
                 (size_t)B_ * H_ * sizeof(float), hipMemcpyDeviceToDevice, stream);
}